// SpectralDecompositionLayer_4028679324129
// MI455X (gfx1250) — compile-verified
//
#include <hip/hip_runtime.h>
#include <hip/hip_bf16.h>

// Batched 64x64 fp32 symmetric eigendecomposition (eigh) for MI455X / gfx1250.
// Parallel cyclic two-sided Jacobi in LDS; eigenvector accumulation V <- V*J
// runs on the matrix pipe via v_wmma_f32_16x16x4_f32 (the only fp32 WMMA).

typedef __attribute__((ext_vector_type(2))) float v2f;
typedef __attribute__((ext_vector_type(8))) float v8f;

namespace {
constexpr int kN       = 64;    // matrix dimension
constexpr int kLda     = 65;    // padded LDS stride (64 banks -> conflict-free cols)
constexpr int kPairs   = 32;    // disjoint rotations per round
constexpr int kThreads = 128;   // 4 wave32 per matrix
constexpr int kSweeps  = 8;     // fixed sweep count (deterministic)
}

__global__ __launch_bounds__(kThreads)
void eigh64_jacobi_wmma(const float* __restrict__ cov,
                        float* __restrict__ out_vals,
                        float* __restrict__ out_vecs)
{
    __shared__ float A[kN * kLda];      // working symmetric matrix
    __shared__ float V[kN * kLda];      // accumulated eigenvectors (columns)
    __shared__ float J[kN * kLda];      // dense per-round rotation matrix
    __shared__ float rc[kPairs], rs[kPairs];
    __shared__ int   pp[kPairs], qq[kPairs];
    __shared__ float diag[kN];
    __shared__ int   rank[kN];

    const int b   = blockIdx.x;
    const int tid = threadIdx.x;
    const float* __restrict__ src = cov + (size_t)b * kN * kN;

    // ---- load A, init V = I ----
    for (int idx = tid; idx < kN * kN; idx += kThreads) {
        const int r = idx >> 6, c = idx & 63;
        A[r * kLda + c] = src[idx];
        V[r * kLda + c] = (r == c) ? 1.0f : 0.0f;
    }
    __syncthreads();

    const int wave  = tid >> 5;        // 0..3
    const int lane  = tid & 31;
    const int lrow  = lane & 15;       // M (A frag) / N (B frag, C store) index
    const int khalf = lane >> 4;       // 0 or 1: selects K sub-pair / M upper half

    for (int sweep = 0; sweep < kSweeps; ++sweep) {
        for (int round = 0; round < kN - 1; ++round) {
            // ---- tournament pairing + rotation angles (threads 0..31) ----
            if (tid < kPairs) {
                const int k = tid;
                auto player = [&](int s) -> int {
                    return (s == 0) ? 0 : 1 + (s - 1 + round) % (kN - 1);
                };
                int p = player(k);
                int q = player(kN - 1 - k);
                if (p > q) { int t = p; p = q; q = t; }
                pp[k] = p; qq[k] = q;

                const float app = A[p * kLda + p];
                const float aqq = A[q * kLda + q];
                const float apq = A[p * kLda + q];
                float c, s;
                if (__builtin_fabsf(apq) < 1e-30f) {
                    c = 1.0f; s = 0.0f;
                } else {
                    const float tau = (aqq - app) / (2.0f * apq);
                    const float t = ((tau >= 0.0f) ? 1.0f : -1.0f) /
                                    (__builtin_fabsf(tau) + __builtin_sqrtf(1.0f + tau * tau));
                    c = __frsqrt_rn(1.0f + t * t);
                    s = t * c;
                }
                rc[k] = c; rs[k] = s;
            }
            __syncthreads();

            // ---- A <- J^T A (row rotations), pairs are disjoint ----
            for (int w = tid; w < kPairs * kN; w += kThreads) {
                const int k = w >> 6, j = w & 63;
                const int p = pp[k], q = qq[k];
                const float c = rc[k], s = rs[k];
                const float ap = A[p * kLda + j];
                const float aq = A[q * kLda + j];
                A[p * kLda + j] = c * ap - s * aq;
                A[q * kLda + j] = s * ap + c * aq;
            }
            __syncthreads();

            // ---- A <- A J (column rotations) ; also rebuild J = I ----
            for (int w = tid; w < kPairs * kN; w += kThreads) {
                const int k = w >> 6, i = w & 63;
                const int p = pp[k], q = qq[k];
                const float c = rc[k], s = rs[k];
                const float ip = A[i * kLda + p];
                const float iq = A[i * kLda + q];
                A[i * kLda + p] = c * ip - s * iq;
                A[i * kLda + q] = s * ip + c * iq;
            }
            for (int idx = tid; idx < kN * kN; idx += kThreads) {
                const int r = idx >> 6, c = idx & 63;
                J[r * kLda + c] = (r == c) ? 1.0f : 0.0f;
            }
            __syncthreads();
            if (tid < kPairs) {
                const int p = pp[tid], q = qq[tid];
                const float c = rc[tid], s = rs[tid];
                J[p * kLda + p] = c;  J[q * kLda + q] = c;
                J[p * kLda + q] = s;  J[q * kLda + p] = -s;
            }
            __syncthreads();

            // ---- V <- V * J on the matrix pipe: 4x4 grid of 16x16 tiles,
            //      K accumulated in steps of 4 via v_wmma_f32_16x16x4_f32.
            //      All tiles land in registers first, then written back in place.
            v8f acc[4];
            #pragma unroll
            for (int t = 0; t < 4; ++t) {
                const int tile = wave + 4 * t;       // 0..15
                const int mt = tile >> 2, nt = tile & 3;
                v8f c = {};
                #pragma unroll
                for (int k0 = 0; k0 < kN; k0 += 4) {
                    // A fragment: V[mt*16+lrow][k0 + 2*khalf + {0,1}] (row-contiguous)
                    const float* ar = &V[(mt * 16 + lrow) * kLda + k0 + 2 * khalf];
                    v2f af; af[0] = ar[0]; af[1] = ar[1];
                    // B fragment: J[k0 + 2*khalf + v][nt*16 + lrow]
                    v2f bf;
                    bf[0] = J[(k0 + 2 * khalf + 0) * kLda + nt * 16 + lrow];
                    bf[1] = J[(k0 + 2 * khalf + 1) * kLda + nt * 16 + lrow];
                    c = __builtin_amdgcn_wmma_f32_16x16x4_f32(
                            false, af, false, bf, (short)0, c, false, false);
                }
                acc[t] = c;
            }
            __syncthreads();   // all reads of V complete
            #pragma unroll
            for (int t = 0; t < 4; ++t) {
                const int tile = wave + 4 * t;
                const int mt = tile >> 2, nt = tile & 3;
                #pragma unroll
                for (int r = 0; r < 8; ++r) {
                    V[(mt * 16 + r + 8 * khalf) * kLda + nt * 16 + lrow] = acc[t][r];
                }
            }
            __syncthreads();
        }
    }

    // ---- sort eigenvalues ascending (rank sort, deterministic tie-break) ----
    if (tid < kN) diag[tid] = A[tid * kLda + tid];
    __syncthreads();
    if (tid < kN) {
        const float d = diag[tid];
        int r = 0;
        for (int j = 0; j < kN; ++j) {
            const float dj = diag[j];
            if (dj < d || (dj == d && j < tid)) ++r;
        }
        rank[tid] = r;
        out_vals[(size_t)b * kN + r] = d;   // (B, N, 1) flat
    }
    __syncthreads();

    // ---- scatter eigenvector columns into ascending order: out[b][i][rank[j]] = V[i][j]
    float* __restrict__ ov = out_vecs + (size_t)b * kN * kN;
    for (int idx = tid; idx < kN * kN; idx += kThreads) {
        const int i = idx >> 6, j = idx & 63;
        ov[i * kN + rank[j]] = V[i * kLda + j];
    }
}

extern "C" void kernel_launch(void* const* d_in, const int* in_sizes, int n_in,
                              void* d_out, int out_size, void* d_ws, size_t ws_size,
                              hipStream_t stream) {
    (void)in_sizes; (void)n_in; (void)d_ws; (void)ws_size; (void)out_size;
    const float* cov = (const float*)d_in[0];
    float* out = (float*)d_out;
    const int B = 8192;
    // d_out layout: eigenvalues (B*64) then eigenvectors (B*64*64), flat in return order.
    float* out_vals = out;
    float* out_vecs = out + (size_t)B * kN;
    eigh64_jacobi_wmma<<<B, kThreads, 0, stream>>>(cov, out_vals, out_vecs);
}